// PCloudConv3d_54640573939789
// MI455X (gfx1250) — compile-verified
//
#include <hip/hip_runtime.h>
#include <hip/hip_bf16.h>

#define N_POINTS  100000
#define DEGREE    32
#define C_IN      64
#define C_OUT     64
#define NUM_BASIS 27
#define BN_EPS    0.001f

typedef float v8f __attribute__((ext_vector_type(8)));
typedef float v2f __attribute__((ext_vector_type(2)));

// LDS strides padded to 68 floats so 32-lane float2 accesses hit all 64 banks.
#define FSTRIDE 68

// ---------------------------------------------------------------------------
// Kernel 1: fused edge-gather aggregation + 64x64 GEMM (fp32 WMMA) + bias+ReLU
// One block = 32 points, 256 threads = 8 waves.
// ---------------------------------------------------------------------------
__global__ __launch_bounds__(256)
void pconv_fused_kernel(const float* __restrict__ inputs,         // [N,64]
                        const float* __restrict__ spatial_w,      // [27,64]
                        const float* __restrict__ depth_w,        // [64,64] (k-major)
                        const float* __restrict__ biases,         // [64]
                        const int*   __restrict__ nn_count,       // [N]
                        const int*   __restrict__ nn_index,       // [N*32,2]
                        const int*   __restrict__ filt_index,     // [N*32]
                        float*       __restrict__ out)            // [N,64]
{
    __shared__ float swLds[NUM_BASIS * C_IN];      // 6912 B
    __shared__ float featLds[32 * FSTRIDE];        // 8704 B
    __shared__ float WtLds[C_OUT * FSTRIDE];       // 17408 B  (transposed: [n][k])

    const int tid  = threadIdx.x;
    const int lane = tid & 31;
    const int wave = __builtin_amdgcn_readfirstlane(tid >> 5);  // force scalar
    const int blockBase = blockIdx.x * 32;

    // Preload spatial weights and transposed depth weights into LDS.
    for (int i = tid; i < NUM_BASIS * C_IN; i += 256)
        swLds[i] = spatial_w[i];
    for (int i = tid; i < C_IN * C_OUT; i += 256) {
        int k = i >> 6, n = i & 63;                 // depth_w[k][n]
        WtLds[n * FSTRIDE + k] = depth_w[i];
    }
    __syncthreads();

    // ---- Aggregation: each wave handles 4 points; lane owns channels 2L,2L+1.
    for (int i = 0; i < 4; ++i) {
        const int pl = wave * 4 + i;                // local point 0..31 (uniform)
        const int p  = blockBase + pl;
        const int eb = p * DEGREE;                  // uniform -> scalar loads
        v2f acc = {0.f, 0.f};
        #pragma unroll 4
        for (int e = 0; e < DEGREE; ++e) {
            const int src = nn_index[(size_t)(eb + e) * 2 + 1];
            const int f   = filt_index[eb + e];
            v2f x = *(const v2f*)(inputs + (size_t)src * C_IN + 2 * lane);
            v2f w = *(const v2f*)(&swLds[f * C_IN + 2 * lane]);
            acc += x * w;
        }
        const float inv = 1.0f / (float)nn_count[p];
        acc *= inv;
        *(v2f*)(&featLds[pl * FSTRIDE + 2 * lane]) = acc;
    }
    __syncthreads();

    // ---- GEMM: wave -> one 16x16 tile of out[32x64] = feat[32x64] @ W[64x64].
    const int Mtile = wave >> 2;
    const int Ntile = wave & 3;
    const int mrow  = Mtile * 16 + (lane & 15);
    const int nrow  = Ntile * 16 + (lane & 15);
    const int koff  = 2 * (lane >> 4);              // ISA A/B frag: hi half -> K+2

    v8f c = {};
    #pragma unroll
    for (int k0 = 0; k0 < 16; ++k0) {               // K = 64 = 16 x 4
        v2f a = *(const v2f*)(&featLds[mrow * FSTRIDE + k0 * 4 + koff]);
        v2f b = *(const v2f*)(&WtLds[nrow * FSTRIDE + k0 * 4 + koff]);
        c = __builtin_amdgcn_wmma_f32_16x16x4_f32(false, a, false, b,
                                                  (short)0, c, false, false);
    }

    // ---- Epilogue: bias + ReLU, store. C layout: VGPR i -> row i + 8*(lane>=16).
    const int col   = Ntile * 16 + (lane & 15);
    const float bn  = biases[col];
    const int rbase = blockBase + Mtile * 16 + ((lane >> 4) << 3);
    #pragma unroll
    for (int i = 0; i < 8; ++i) {
        float v = c[i] + bn;
        v = v > 0.f ? v : 0.f;
        out[(size_t)(rbase + i) * C_OUT + col] = v;
    }
}

// ---------------------------------------------------------------------------
// Kernel 2: per-channel sum / sum-of-squares (deterministic tree reduction).
// 64 blocks, one channel each. Reads hit L2 (out was just written, 25.6 MB).
// ---------------------------------------------------------------------------
__global__ __launch_bounds__(256)
void channel_stats_kernel(const float* __restrict__ out, float* __restrict__ ws)
{
    const int ch = blockIdx.x;
    const int t  = threadIdx.x;
    float s = 0.f, s2 = 0.f;
    for (int r = t; r < N_POINTS; r += 256) {
        float v = out[(size_t)r * C_OUT + ch];
        s += v; s2 += v * v;
    }
    __shared__ float ls[256], ls2[256];
    ls[t] = s; ls2[t] = s2;
    __syncthreads();
    for (int off = 128; off > 0; off >>= 1) {
        if (t < off) { ls[t] += ls[t + off]; ls2[t] += ls2[t + off]; }
        __syncthreads();
    }
    if (t == 0) { ws[ch] = ls[0]; ws[C_OUT + ch] = ls2[0]; }
}

// ---------------------------------------------------------------------------
// Kernel 3: apply batch-norm in place. One float2 (two channels) per thread.
// ---------------------------------------------------------------------------
__global__ __launch_bounds__(256)
void bn_apply_kernel(float* __restrict__ out, const float* __restrict__ ws,
                     const float* __restrict__ gamma,
                     const float* __restrict__ beta)
{
    const size_t idx = (size_t)blockIdx.x * blockDim.x + threadIdx.x;
    const size_t base = idx * 2;
    if (base >= (size_t)N_POINTS * C_OUT) return;
    const int n = (int)(base & 63);
    const float invN = 1.0f / (float)N_POINTS;

    float m0 = ws[n]     * invN;
    float m1 = ws[n + 1] * invN;
    float v0 = ws[C_OUT + n]     * invN - m0 * m0;
    float v1 = ws[C_OUT + n + 1] * invN - m1 * m1;
    float sc0 = __frsqrt_rn(v0 + BN_EPS) * gamma[n];
    float sc1 = __frsqrt_rn(v1 + BN_EPS) * gamma[n + 1];

    v2f o = *(v2f*)(out + base);
    o.x = (o.x - m0) * sc0 + beta[n];
    o.y = (o.y - m1) * sc1 + beta[n + 1];
    *(v2f*)(out + base) = o;
}

// ---------------------------------------------------------------------------
extern "C" void kernel_launch(void* const* d_in, const int* in_sizes, int n_in,
                              void* d_out, int out_size, void* d_ws, size_t ws_size,
                              hipStream_t stream)
{
    const float* inputs     = (const float*)d_in[0];
    const float* spatial_w  = (const float*)d_in[1];
    const float* depth_w    = (const float*)d_in[2];
    const float* biases     = (const float*)d_in[3];
    const float* bn_gamma   = (const float*)d_in[4];
    const float* bn_beta    = (const float*)d_in[5];
    const int*   nn_count   = (const int*)d_in[6];
    const int*   nn_index   = (const int*)d_in[7];
    const int*   filt_index = (const int*)d_in[8];
    float*       out        = (float*)d_out;
    float*       ws         = (float*)d_ws;

    pconv_fused_kernel<<<N_POINTS / 32, 256, 0, stream>>>(
        inputs, spatial_w, depth_w, biases, nn_count, nn_index, filt_index, out);

    channel_stats_kernel<<<C_OUT, 256, 0, stream>>>(out, ws);

    const size_t pairs = (size_t)N_POINTS * C_OUT / 2;  // 3.2M
    bn_apply_kernel<<<(int)((pairs + 255) / 256), 256, 0, stream>>>(
        out, ws, bn_gamma, bn_beta);
}